// EdgeEncoding_72894184947752
// MI455X (gfx1250) — compile-verified
//
#include <hip/hip_runtime.h>

typedef __attribute__((ext_vector_type(2))) float v2f;
typedef __attribute__((ext_vector_type(8))) float v8f;

#define EDGE_DIM 64
#define LPATH    5

// ---------------------------------------------------------------------------
// Kernel 1: dots[e][l] = sum_d edge_attr[e][d] * edge_vector[l][d]
// One wave (32 lanes) computes one 16-row tile of edge_attr against the
// zero-padded 64x16 B matrix (columns 0..4 = edge_vector rows, 5..15 = 0)
// via 16 chained V_WMMA_F32_16X16X4_F32 ops (K=64 in chunks of 4, fp32 exact).
// ---------------------------------------------------------------------------
__global__ void __launch_bounds__(256)
dots_wmma_kernel(const float* __restrict__ edge_attr,
                 const float* __restrict__ edge_vector,
                 float* __restrict__ dots,
                 int num_tiles)
{
    const int lane = threadIdx.x & 31;
    const int wave = (int)((blockIdx.x * blockDim.x + threadIdx.x) >> 5);
    if (wave >= num_tiles) return;          // wave-uniform: EXEC stays all-1s

    const int n  = lane & 15;               // col for B/D, row for A
    const int hi = lane >> 4;               // half-wave select
    const int kb = hi * 2;                  // K sub-offset within a K=4 chunk

    // B fragment: B[k][n] = edge_vector[n][k] for n<5, else 0.
    // Lane holds (k0+kb, k0+kb+1) for each of the 16 K-chunks.
    v2f bfrag[16];
#pragma unroll
    for (int c = 0; c < 16; ++c) {
        const int k = c * 4 + kb;
        if (n < LPATH) {
            bfrag[c].x = edge_vector[n * EDGE_DIM + k];
            bfrag[c].y = edge_vector[n * EDGE_DIM + k + 1];
        } else {
            bfrag[c].x = 0.0f;
            bfrag[c].y = 0.0f;
        }
    }

    // A fragment source: row (tile*16 + n), elements k0+kb, k0+kb+1 (8B aligned)
    const float* arow = edge_attr + ((size_t)wave * 16 + n) * EDGE_DIM + kb;

    v8f acc = {};
#pragma unroll
    for (int c = 0; c < 16; ++c) {
        const float2 af = *(const float2*)(arow + c * 4);
        v2f a; a.x = af.x; a.y = af.y;
        acc = __builtin_amdgcn_wmma_f32_16x16x4_f32(
            /*neg_a=*/false, a, /*neg_b=*/false, bfrag[c],
            /*c_mod=*/(short)0, acc, /*reuse_a=*/false, /*reuse_b=*/false);
    }

    // D layout: VGPR r -> row (r + 8*hi), col n. Store only the 5 real cols.
    if (n < LPATH) {
#pragma unroll
        for (int r = 0; r < 8; ++r) {
            const int row = wave * 16 + r + 8 * hi;
            dots[(size_t)row * LPATH + n] = acc[r];
        }
    }
}

// ---------------------------------------------------------------------------
// Kernel 2: memory-bound gather + masked average + nan_to_num.
// One thread per (i,j) pair: streams 5 indices + length (coalesced),
// <=5 random 4B reads of dots (L2-resident, ~1.3MB), one 4B store.
// ---------------------------------------------------------------------------
__global__ void __launch_bounds__(256)
gather_avg_kernel(const int* __restrict__ paths,
                  const int* __restrict__ lens,
                  const float* __restrict__ dots,
                  float* __restrict__ out,
                  int total)
{
    const int t = (int)(blockIdx.x * blockDim.x + threadIdx.x);
    if (t >= total) return;

    const int* p = paths + (size_t)t * LPATH;
    float s = 0.0f;
#pragma unroll
    for (int q = 0; q < LPATH; ++q) {
        const int e = p[q];
        if (e >= 0) s += dots[(size_t)e * LPATH + q];
    }

    const int len = lens[t];
    float r = 0.0f;
    if (len > 0) r = s / ((float)len + 1e-10f);

    // nan_to_num: NaN -> 0, +/-inf -> +/-FLT_MAX
    r = (r != r) ? 0.0f : r;
    const float FMAXV = 3.402823466e38f;
    r = fminf(fmaxf(r, -FMAXV), FMAXV);

    out[t] = r;
}

// ---------------------------------------------------------------------------
// Launch. Input order (setup_inputs): x, edge_attr, edge_vector,
// edge_paths_tensor, edge_paths_length. `x` is unused by the reference.
// ---------------------------------------------------------------------------
extern "C" void kernel_launch(void* const* d_in, const int* in_sizes, int n_in,
                              void* d_out, int out_size, void* d_ws, size_t ws_size,
                              hipStream_t stream)
{
    const float* edge_attr   = (const float*)d_in[1];
    const float* edge_vector = (const float*)d_in[2];
    const int*   paths       = (const int*)d_in[3];
    const int*   lens        = (const int*)d_in[4];
    float*       out         = (float*)d_out;
    float*       dots        = (float*)d_ws;   // E * LPATH floats (~1.3 MB)

    const int E     = in_sizes[1] / EDGE_DIM;  // 65536
    const int tiles = E / 16;                  // 4096 (one wave each)
    const int total = in_sizes[4];             // N*N = 4194304

    const int threadsA = tiles * 32;
    dots_wmma_kernel<<<(threadsA + 255) / 256, 256, 0, stream>>>(
        edge_attr, edge_vector, dots, tiles);

    gather_avg_kernel<<<(total + 255) / 256, 256, 0, stream>>>(
        paths, lens, dots, out, total);
}